// GIN_91036126806159
// MI455X (gfx1250) — compile-verified
//
#include <hip/hip_runtime.h>

typedef __attribute__((ext_vector_type(2))) float v2f;
typedef __attribute__((ext_vector_type(4))) float v4f;
typedef __attribute__((ext_vector_type(8))) float v8f;

#define NNODES 50000
#define NEDGES 800000
#define HID    128
#define HID2   256

// ---------------------------------------------------------------------------
// Zero the aggregation scratch (harness poisons ws; must re-zero every call).
// ---------------------------------------------------------------------------
__global__ void zero_f4(v4f* __restrict__ p, int n4) {
    int i = blockIdx.x * blockDim.x + threadIdx.x;
    if (i < n4) {
        v4f z = {0.f, 0.f, 0.f, 0.f};
        p[i] = z;
    }
}

// ---------------------------------------------------------------------------
// Edge phase: msg = node_feat[src] + edge_feat[e]; atomic scatter-sum to dst.
// One wave32 per edge; each lane owns 4 consecutive floats (128 = 32*4).
// node_feat (25.6MB) and agg (25.6MB) are L2-resident; edge_feat streams.
// ---------------------------------------------------------------------------
__global__ void edge_scatter(const float* __restrict__ node_feat,
                             const float* __restrict__ edge_feat,
                             const int*   __restrict__ esrc,
                             const int*   __restrict__ edst,
                             float*       __restrict__ agg) {
    long gid  = (long)blockIdx.x * blockDim.x + threadIdx.x;
    int  edge = (int)(gid >> 5);
    int  lane = threadIdx.x & 31;
    if (edge >= NEDGES) return;

    int src = esrc[edge];
    int dst = edst[edge];

    v4f nf = *(const v4f*)(node_feat + (size_t)src * HID + lane * 4);
    v4f ef = __builtin_nontemporal_load(
                 (const v4f*)(edge_feat + (size_t)edge * HID + lane * 4));
    v4f m = nf + ef;

    float* a = agg + (size_t)dst * HID + lane * 4;
    atomicAdd(a + 0, m.x);
    atomicAdd(a + 1, m.y);
    atomicAdd(a + 2, m.z);
    atomicAdd(a + 3, m.w);
}

// ---------------------------------------------------------------------------
// Fused GEMM + bias (+ ReLU):  C[16-row tile, NCOLS] = act(A @ B + bias)
// Exact fp32 via V_WMMA_F32_16X16X4_F32 (matches reference precision).
//
// Block = 256 threads = 8 wave32; block owns one 16-row M-tile (A staged in
// LDS via flat float4 copy, read from HBM exactly once). Each wave owns
// NCOLS/128 N-tiles of width 16, accumulated simultaneously inside the K-loop
// so every A-fragment LDS read feeds NTW WMMAs.
//
// Grid is exactly M/16 blocks (M = 50000 is divisible by 16) -> no tail
// guards anywhere; EXEC stays all-ones through every WMMA and every store.
//
// Fragment layouts per CDNA5 ISA 7.12.2:
//   A 16x4 : lane m=lane%16; v[0]=A[m][k+koff], v[1]=A[m][k+koff+1],
//            koff = (lane<16) ? 0 : 2
//   B 4x16 : lane n=lane%16; v[0]=B[k+koff][n], v[1]=B[k+koff+1][n]
//   C/D    : 8 VGPRs; row = v + (lane<16 ? 0 : 8), col = lane%16
// ---------------------------------------------------------------------------
template<int K, int NCOLS, bool RELU, bool NTSTORE>
__global__ __launch_bounds__(256)
void gemm_bias_wmma(const float* __restrict__ A,
                    const float* __restrict__ B,
                    const float* __restrict__ bias,
                    float*       __restrict__ C) {
    __shared__ float As[16 * K];   // 8KB (K=128) or 16KB (K=256)

    const int mbase = blockIdx.x * 16;
    const int tid   = threadIdx.x;

    // Flat vectorized cooperative A-tile load (tile is contiguous in memory).
    constexpr int NV4 = 16 * K / 4;                       // 512 or 1024
    const v4f* Ag  = (const v4f*)(A + (size_t)mbase * K); // 16B aligned
    v4f*       Asv = (v4f*)As;
    #pragma unroll
    for (int i = tid; i < NV4; i += 256) Asv[i] = Ag[i];
    __syncthreads();

    const int wave   = tid >> 5;
    const int lane   = tid & 31;
    const int mn     = lane & 15;               // A-row / B-col within frag
    const int koff   = (lane < 16) ? 0 : 2;
    const int rowoff = (lane < 16) ? 0 : 8;

    constexpr int NTW = NCOLS / 16 / 8;         // N-tiles per wave (2 or 1)
    const int nbase0  = wave * NTW * 16;

    v8f acc[NTW];
    #pragma unroll
    for (int t = 0; t < NTW; ++t) acc[t] = (v8f){};

    #pragma unroll 8
    for (int k = 0; k < K; k += 4) {
        // A fragment: two consecutive K values -> one 8B LDS read, reused
        // across all NTW tiles.
        v2f a = *(const v2f*)(&As[mn * K + k + koff]);
        #pragma unroll
        for (int t = 0; t < NTW; ++t) {
            const int nb = nbase0 + t * 16;
            v2f b;                               // L2-resident weights
            b.x = B[(size_t)(k + koff)     * NCOLS + nb + mn];
            b.y = B[(size_t)(k + koff + 1) * NCOLS + nb + mn];
            acc[t] = __builtin_amdgcn_wmma_f32_16x16x4_f32(
                         /*neg_a=*/false, a, /*neg_b=*/false, b,
                         /*c_mod=*/(short)0, acc[t],
                         /*reuse_a=*/false, /*reuse_b=*/false);
        }
    }

    // Straight-line epilogue: bias (+ReLU) and store, no guards needed.
    #pragma unroll
    for (int t = 0; t < NTW; ++t) {
        const int nb = nbase0 + t * 16;
        const float bv = bias[nb + mn];
        #pragma unroll
        for (int v = 0; v < 8; ++v) {
            const int m = mbase + rowoff + v;
            float val = acc[t][v] + bv;
            if (RELU) val = fmaxf(val, 0.0f);
            float* dst = C + (size_t)m * NCOLS + nb + mn;
            if (NTSTORE) __builtin_nontemporal_store(val, dst);
            else         *dst = val;
        }
    }
}

// ---------------------------------------------------------------------------
// Launch: zero agg -> edge scatter -> layer1 (ReLU) -> layer2
// ws layout: agg [NNODES*HID] f32, then h [NNODES*HID2] f32 (76.8MB total)
// ---------------------------------------------------------------------------
extern "C" void kernel_launch(void* const* d_in, const int* in_sizes, int n_in,
                              void* d_out, int out_size, void* d_ws, size_t ws_size,
                              hipStream_t stream) {
    const float* node_feat = (const float*)d_in[0];
    const float* edge_feat = (const float*)d_in[1];
    const int*   esrc      = (const int*)  d_in[2];
    const int*   edst      = (const int*)  d_in[3];
    const float* w1        = (const float*)d_in[4];
    const float* b1        = (const float*)d_in[5];
    const float* w2        = (const float*)d_in[6];
    const float* b2        = (const float*)d_in[7];
    float* out = (float*)d_out;

    float* agg = (float*)d_ws;                       // [NNODES, HID]
    float* h   = agg + (size_t)NNODES * HID;         // [NNODES, HID2]

    // 1) zero agg
    int n4 = NNODES * HID / 4;
    zero_f4<<<(n4 + 255) / 256, 256, 0, stream>>>((v4f*)agg, n4);

    // 2) edge gather + scatter-add (one wave per edge)
    long ethreads = (long)NEDGES * 32;
    edge_scatter<<<(int)((ethreads + 255) / 256), 256, 0, stream>>>(
        node_feat, edge_feat, esrc, edst, agg);

    // 3) h = relu(agg @ w1 + b1)    [50000,128]x[128,256], h stays hot in L2
    gemm_bias_wmma<HID, HID2, true, false><<<NNODES / 16, 256, 0, stream>>>(
        agg, w1, b1, h);

    // 4) out = h @ w2 + b2          [50000,256]x[256,128], NT store to d_out
    gemm_bias_wmma<HID2, HID, false, true><<<NNODES / 16, 256, 0, stream>>>(
        h, w2, b2, out);
}